// APPNPNet_62423054680287
// MI455X (gfx1250) — compile-verified
//
#include <hip/hip_runtime.h>
#include <hip/hip_bf16.h>
#include <math.h>

#define NNODES 100000
#define NEDGES 3200000
#define FIN    512
#define FHID   16
#define FOUT   40
#define KPROP  10
#define ALPHA  0.1f

typedef __attribute__((ext_vector_type(2))) float v2f;
typedef __attribute__((ext_vector_type(8))) float v8f;

// ---------------------------------------------------------------------------
// Degree / normalization kernels
// ---------------------------------------------------------------------------
__global__ void deg_init_kernel(float* __restrict__ deg) {
    int i = blockIdx.x * blockDim.x + threadIdx.x;
    if (i < NNODES) deg[i] = 1.0f;  // self-loop contributes 1
}

__global__ void deg_count_kernel(const int* __restrict__ dst, float* __restrict__ deg) {
    int e = blockIdx.x * blockDim.x + threadIdx.x;
    if (e < NEDGES) atomicAdd(&deg[dst[e]], 1.0f);
}

__global__ void deg_finalize_kernel(float* __restrict__ dinv) {
    int i = blockIdx.x * blockDim.x + threadIdx.x;
    if (i < NNODES) dinv[i] = rsqrtf(dinv[i]);   // deg >= 1 always (self loop)
}

// ---------------------------------------------------------------------------
// Fused MLP: H0 = (relu(x @ W1^T + b1)) @ W2^T + b2, via fp32 WMMA 16x16x4.
// One wave per 16-node tile. GEMM1: K=512 -> 128 WMMAs. Result staged through
// LDS (C/D layout -> A layout), then GEMM2 over 3 column chunks of 16.
// ---------------------------------------------------------------------------
__global__ __launch_bounds__(256) void mlp_wmma_kernel(
        const float* __restrict__ x,
        const float* __restrict__ W1, const float* __restrict__ b1,
        const float* __restrict__ W2, const float* __restrict__ b2,
        float* __restrict__ h0, float* __restrict__ h)
{
    __shared__ float lds[8][FHID * FHID];   // 1 KB per wave

    const int wave = threadIdx.x >> 5;
    const int lane = threadIdx.x & 31;
    const int tile = blockIdx.x * 8 + wave;
    if (tile >= (NNODES / 16)) return;      // uniform per wave -> EXEC all-ones inside

    const int row0 = tile * 16;
    const int lm = lane & 15;               // M row (A) / N col (B,C,D)
    const int lh = lane >> 4;               // K-half selector

    // ---- GEMM1: A = x tile [16 x 512], B = W1^T chunk [4 x 16] per step ----
    const float* xrow = x + (size_t)(row0 + lm) * FIN;
    const float* wrow = W1 + (size_t)lm * FIN;
    v8f acc = {};
    for (int k0 = 0; k0 < FIN; k0 += 4) {
        const int k = k0 + 2 * lh;
        v2f a; a[0] = xrow[k]; a[1] = xrow[k + 1];   // A: lane lm holds (M=lm, K=k..k+1)
        v2f b; b[0] = wrow[k]; b[1] = wrow[k + 1];   // B: lane lm holds (K=k..k+1, N=lm)
        acc = __builtin_amdgcn_wmma_f32_16x16x4_f32(
                false, a, false, b, (short)0, acc, false, false);
    }

    // ---- bias + ReLU, spill C/D layout to LDS as row-major 16x16 ----
    const float bias1 = b1[lm];
    float* L = &lds[wave][0];
#pragma unroll
    for (int g = 0; g < 8; ++g) {
        const float v = fmaxf(acc[g] + bias1, 0.0f);   // VGPR g: M = g + 8*lh, N = lm
        L[(g + 8 * lh) * FHID + lm] = v;
    }
    // intra-wave LDS is in-order (DScnt), no barrier needed before re-read

    // ---- GEMM2: A = H1 [16 x 16] from LDS, B = W2^T [16 x 16-chunk] ----
    for (int nc = 0; nc < 48; nc += 16) {
        const int nn = nc + lm;             // output column
        v8f acc2 = {};
#pragma unroll
        for (int k0 = 0; k0 < FHID; k0 += 4) {
            const int k = k0 + 2 * lh;
            v2f a; a[0] = L[lm * FHID + k]; a[1] = L[lm * FHID + k + 1];
            v2f b;
            if (nn < FOUT) { b[0] = W2[nn * FHID + k]; b[1] = W2[nn * FHID + k + 1]; }
            else           { b[0] = 0.0f;              b[1] = 0.0f; }
            acc2 = __builtin_amdgcn_wmma_f32_16x16x4_f32(
                    false, a, false, b, (short)0, acc2, false, false);
        }
        if (nn < FOUT) {
            const float bias2 = b2[nn];
#pragma unroll
            for (int g = 0; g < 8; ++g) {
                const int r = row0 + g + 8 * lh;
                const float v = acc2[g] + bias2;
                h0[(size_t)r * FOUT + nn] = v;
                h [(size_t)r * FOUT + nn] = v;
            }
        }
    }
}

// ---------------------------------------------------------------------------
// Propagation kernels (h, agg resident in 192MB L2: 16MB each)
// ---------------------------------------------------------------------------
// agg := h * dinv^2   (self-loop term), float4 over N*40 contiguous floats
__global__ void agg_init_kernel(const float* __restrict__ h,
                                const float* __restrict__ dinv,
                                float* __restrict__ agg) {
    int i = blockIdx.x * blockDim.x + threadIdx.x;   // over N*10 float4s
    if (i >= NNODES * (FOUT / 4)) return;
    const float di = dinv[i / (FOUT / 4)];
    const float w = di * di;
    float4 v = ((const float4*)h)[i];
    ((float4*)agg)[i] = make_float4(v.x * w, v.y * w, v.z * w, v.w * w);
}

// agg[dst] += h[src] * dinv[src]*dinv[dst] for every edge (L2-resident atomics)
__global__ void edge_scatter_kernel(const int* __restrict__ src,
                                    const int* __restrict__ dst,
                                    const float* __restrict__ dinv,
                                    const float* __restrict__ h,
                                    float* __restrict__ agg) {
    int e = blockIdx.x * blockDim.x + threadIdx.x;
    if (e >= NEDGES) return;
    const int s = src[e];
    const int d = dst[e];
    const float nrm = dinv[s] * dinv[d];
    const float4* hs = (const float4*)(h + (size_t)s * FOUT);
    float* ad = agg + (size_t)d * FOUT;
#pragma unroll
    for (int q = 0; q < FOUT / 4; ++q) {
        float4 v = hs[q];
        atomicAdd(&ad[4 * q + 0], v.x * nrm);
        atomicAdd(&ad[4 * q + 1], v.y * nrm);
        atomicAdd(&ad[4 * q + 2], v.z * nrm);
        atomicAdd(&ad[4 * q + 3], v.w * nrm);
    }
}

// h := (1-alpha)*agg + alpha*h0
__global__ void blend_kernel(const float* __restrict__ agg,
                             const float* __restrict__ h0,
                             float* __restrict__ h) {
    int i = blockIdx.x * blockDim.x + threadIdx.x;   // over N*10 float4s
    if (i >= NNODES * (FOUT / 4)) return;
    float4 a = ((const float4*)agg)[i];
    float4 z = ((const float4*)h0)[i];
    ((float4*)h)[i] = make_float4((1.0f - ALPHA) * a.x + ALPHA * z.x,
                                  (1.0f - ALPHA) * a.y + ALPHA * z.y,
                                  (1.0f - ALPHA) * a.z + ALPHA * z.z,
                                  (1.0f - ALPHA) * a.w + ALPHA * z.w);
}

// ---------------------------------------------------------------------------
// Row-wise log-softmax over 40 channels
// ---------------------------------------------------------------------------
__global__ void log_softmax_kernel(const float* __restrict__ h,
                                   float* __restrict__ out) {
    int row = blockIdx.x * blockDim.x + threadIdx.x;
    if (row >= NNODES) return;
    const float* r = h + (size_t)row * FOUT;
    float m = -INFINITY;
#pragma unroll
    for (int c = 0; c < FOUT; ++c) m = fmaxf(m, r[c]);
    float s = 0.0f;
#pragma unroll
    for (int c = 0; c < FOUT; ++c) s += __expf(r[c] - m);
    const float lse = __logf(s) + m;
    float* o = out + (size_t)row * FOUT;
#pragma unroll
    for (int c = 0; c < FOUT; ++c) o[c] = r[c] - lse;
}

// ---------------------------------------------------------------------------
extern "C" void kernel_launch(void* const* d_in, const int* in_sizes, int n_in,
                              void* d_out, int out_size, void* d_ws, size_t ws_size,
                              hipStream_t stream) {
    const float* x  = (const float*)d_in[0];
    const int*   ei = (const int*)d_in[1];      // [2, E] flat: src then dst
    const float* W1 = (const float*)d_in[2];
    const float* b1 = (const float*)d_in[3];
    const float* W2 = (const float*)d_in[4];
    const float* b2 = (const float*)d_in[5];
    float* out = (float*)d_out;

    const int* src = ei;
    const int* dst = ei + NEDGES;

    // workspace layout (floats): h0 | h | agg | dinv  (~48.4 MB)
    float* ws   = (float*)d_ws;
    float* h0   = ws;
    float* h    = h0 + (size_t)NNODES * FOUT;
    float* agg  = h  + (size_t)NNODES * FOUT;
    float* dinv = agg + (size_t)NNODES * FOUT;

    const int B = 256;
    const int gN   = (NNODES + B - 1) / B;
    const int gE   = (NEDGES + B - 1) / B;
    const int gV4  = (NNODES * (FOUT / 4) + B - 1) / B;
    const int gMLP = (NNODES / 16 + 7) / 8;     // 8 waves (tiles) per 256-thread block

    // normalization
    deg_init_kernel<<<gN, B, 0, stream>>>(dinv);
    deg_count_kernel<<<gE, B, 0, stream>>>(dst, dinv);
    deg_finalize_kernel<<<gN, B, 0, stream>>>(dinv);

    // fused 2-layer MLP (WMMA), writes h0 and h
    mlp_wmma_kernel<<<gMLP, B, 0, stream>>>(x, W1, b1, W2, b2, h0, h);

    // K propagation rounds
    for (int t = 0; t < KPROP; ++t) {
        agg_init_kernel<<<gV4, B, 0, stream>>>(h, dinv, agg);
        edge_scatter_kernel<<<gE, B, 0, stream>>>(src, dst, dinv, h, agg);
        blend_kernel<<<gV4, B, 0, stream>>>(agg, h0, h);
    }

    log_softmax_kernel<<<gN, B, 0, stream>>>(h, out);
}